// ScaledDotProductAttention_33681133535371
// MI455X (gfx1250) — compile-verified
//
#include <hip/hip_runtime.h>
#include <math.h>

// ---------------------------------------------------------------------------
// Problem constants (from reference): B=32, QL=16 (only q0 used), KL=8192, D=512
// ---------------------------------------------------------------------------
#define B_   32
#define QL_  16
#define KL_  8192
#define D_   512
#define RSQRT_DIM 0.044194173824159216f   // 1/sqrt(512)

#define KEYS_PER_BLOCK 128                 // score kernel: 8 waves * 16 keys
#define SCORE_CHUNKS   (KL_ / KEYS_PER_BLOCK)   // 64 per batch
#define CTX_KCHUNK     512                 // context kernel keys per block
#define CTX_SPLITS     (KL_ / CTX_KCHUNK)  // 16 per batch

typedef float v2f __attribute__((ext_vector_type(2)));
typedef float v8f __attribute__((ext_vector_type(8)));

// ---------------------------------------------------------------------------
// Kernel 1: scores via V_WMMA_F32_16X16X4_F32.
// grid = (SCORE_CHUNKS, B), block = 256 (8 wave32).
// Wave w owns 16 key rows. A = 16x4 f32 tile of key (documented layout:
//   lane L: row m = L&15, lane-half selects K pair {0,1} / {2,3});
// B = 4x16 with q[d..d+3] broadcast across all 16 columns (mirrored K slots).
// Accumulate over D in 128 steps of K=4. Every column of C holds the same
// dot product for its row; lanes 0 and 16 extract rows 0..7 / 8..15.
// Writes w[b,k] = exp((q.k)/sqrt(D)*t)*t and a per-block partial sum.
// ---------------------------------------------------------------------------
__global__ __launch_bounds__(256)
void score_wmma_kernel(const float* __restrict__ query,
                       const float* __restrict__ key,
                       const float* __restrict__ trust,
                       float* __restrict__ wbuf,
                       float* __restrict__ psums)
{
    const int b     = blockIdx.y;
    const int chunk = blockIdx.x;
    const int kbase = chunk * KEYS_PER_BLOCK;

    const int tid  = threadIdx.x;
    const int wave = tid >> 5;
    const int lane = tid & 31;
    const int m    = lane & 15;     // key row within wave tile
    const int half = lane >> 4;     // selects K pair

    __shared__ float qs[D_];
    __shared__ float wave_partial[16];

    // Stage q0 for this batch into LDS (uniform trip count: no divergence
    // before the WMMA loop, EXEC stays all-ones as WMMA requires).
    const float* q0 = query + (size_t)b * QL_ * D_;   // row 0 of query[b]
    for (int i = tid; i < D_; i += 256) qs[i] = q0[i];
    __syncthreads();

    const float* krow = key + ((size_t)b * KL_ + kbase + wave * 16 + m) * D_ + 2 * half;

    v8f c = {};
    #pragma unroll 8
    for (int d = 0; d < D_; d += 4) {
        v2f a  = *(const v2f*)(krow + d);          // A slots K = 2*half, 2*half+1
        v2f bq = *(const v2f*)(&qs[d + 2 * half]); // B slots mirror A's K mapping
        c = __builtin_amdgcn_wmma_f32_16x16x4_f32(
                /*neg_a=*/false, a, /*neg_b=*/false, bq,
                /*c_mod=*/(short)0, c, /*reuse_a=*/false, /*reuse_b=*/false);
    }

    // C layout: VGPR j, lanes 0-15 -> row j ; lanes 16-31 -> row j+8.
    float local = 0.0f;
    if (m == 0) {   // lanes 0 and 16 of each wave
        const size_t bk = (size_t)b * KL_;
        #pragma unroll
        for (int j = 0; j < 8; ++j) {
            const int k = kbase + wave * 16 + half * 8 + j;
            const float t = trust[bk + k];
            const float s = c[j] * RSQRT_DIM * t;   // (q.k)/sqrt(D) * trust
            const float w = expf(s) * t;            // exp(score) * trust
            wbuf[bk + k] = w;
            local += w;
        }
        wave_partial[wave * 2 + half] = local;
    }
    __syncthreads();

    // Fixed-order block sum -> deterministic.
    if (tid == 0) {
        float s = 0.0f;
        #pragma unroll
        for (int i = 0; i < 16; ++i) s += wave_partial[i];
        psums[b * SCORE_CHUNKS + chunk] = s;
    }
}

// ---------------------------------------------------------------------------
// Kernel 2: reduce per-block partial sums -> sums[b]. grid=B, block=64.
// Fixed-order tree: deterministic.
// ---------------------------------------------------------------------------
__global__ __launch_bounds__(64)
void sum_reduce_kernel(const float* __restrict__ psums,
                       float* __restrict__ sums)
{
    const int b   = blockIdx.x;
    const int tid = threadIdx.x;
    __shared__ float sm[64];
    sm[tid] = psums[b * SCORE_CHUNKS + tid];
    __syncthreads();
    for (int s = 32; s > 0; s >>= 1) {
        if (tid < s) sm[tid] += sm[tid + s];
        __syncthreads();
    }
    if (tid == 0) sums[b] = sm[0];
}

// ---------------------------------------------------------------------------
// Kernel 3: partial context. grid = (CTX_SPLITS, B), block = 256.
// Thread t owns output elements d = 2t, 2t+1. Each value row is read as
// 256 x float2 = 2 KB fully contiguous (perfect coalescing); prefetch ahead.
// ---------------------------------------------------------------------------
__global__ __launch_bounds__(256)
void context_partial_kernel(const float* __restrict__ value,
                            const float* __restrict__ wbuf,
                            float* __restrict__ part)
{
    const int b     = blockIdx.y;
    const int split = blockIdx.x;
    const int kbase = split * CTX_KCHUNK;
    const int tid   = threadIdx.x;
    const int d0    = tid * 2;

    const float* vrow = value + ((size_t)b * KL_ + kbase) * D_ + d0;
    const float* wp   = wbuf + (size_t)b * KL_ + kbase;

    float acc0 = 0.0f, acc1 = 0.0f;
    for (int k = 0; k < CTX_KCHUNK; ++k) {
        const float w = wp[k];                       // broadcast (L2/L0 hit)
        const v2f  vv = *(const v2f*)vrow;
        acc0 += w * vv.x;
        acc1 += w * vv.y;
        vrow += D_;
        __builtin_prefetch(vrow + 4 * D_, 0, 1);     // global_prefetch_b8, speculative
    }

    float* p = part + ((size_t)b * CTX_SPLITS + split) * D_ + d0;
    p[0] = acc0;
    p[1] = acc1;
}

// ---------------------------------------------------------------------------
// Kernel 4: out[b,d] = (1/sums[b]) * sum_splits part[b,split,d].
// grid = B*D/256 = 64, block = 256. Fixed-order loop: deterministic.
// ---------------------------------------------------------------------------
__global__ __launch_bounds__(256)
void final_reduce_kernel(const float* __restrict__ part,
                         const float* __restrict__ sums,
                         float* __restrict__ out)
{
    const int idx = blockIdx.x * 256 + threadIdx.x;   // 0 .. B*D-1
    const int b   = idx >> 9;                         // /512
    const int d   = idx & (D_ - 1);

    const float inv = 1.0f / sums[b];
    float s = 0.0f;
    #pragma unroll
    for (int sp = 0; sp < CTX_SPLITS; ++sp)
        s += part[((size_t)b * CTX_SPLITS + sp) * D_ + d];
    out[(size_t)b * D_ + d] = s * inv;
}

// ---------------------------------------------------------------------------
// Launch: inputs (query, key, value, trust_score) in setup_inputs() order.
// Workspace layout (floats):
//   [0)              wbuf   : B*KL          = 262144
//   [262144)         psums  : B*64          =   2048
//   [264192)         sums   : B             =     32
//   [264224)         part   : B*16*D        = 262144
// total ~2.06 MB.
// ---------------------------------------------------------------------------
extern "C" void kernel_launch(void* const* d_in, const int* in_sizes, int n_in,
                              void* d_out, int out_size, void* d_ws, size_t ws_size,
                              hipStream_t stream)
{
    const float* query = (const float*)d_in[0];
    const float* key   = (const float*)d_in[1];
    const float* value = (const float*)d_in[2];
    const float* trust = (const float*)d_in[3];
    float* out = (float*)d_out;

    float* wbuf  = (float*)d_ws;
    float* psums = wbuf + (size_t)B_ * KL_;
    float* sums  = psums + B_ * SCORE_CHUNKS;
    float* part  = sums + B_;

    (void)in_sizes; (void)n_in; (void)out_size; (void)ws_size;

    score_wmma_kernel<<<dim3(SCORE_CHUNKS, B_), 256, 0, stream>>>(query, key, trust, wbuf, psums);
    sum_reduce_kernel<<<dim3(B_), 64, 0, stream>>>(psums, sums);
    context_partial_kernel<<<dim3(CTX_SPLITS, B_), 256, 0, stream>>>(value, wbuf, part);
    final_reduce_kernel<<<dim3((B_ * D_) / 256), 256, 0, stream>>>(part, sums, out);
}